// GAT_52913997086749
// MI455X (gfx1250) — compile-verified
//
#include <hip/hip_runtime.h>
#include <hip/hip_bf16.h>
#include <math.h>

// Problem constants (match reference)
#define NN   50000
#define RR   4
#define EMB  128
#define HH   4
#define SS   32
#define EE   400000

typedef float v2f __attribute__((ext_vector_type(2)));
typedef float v8f __attribute__((ext_vector_type(8)));

// ---- order-preserving float<->uint encoding for atomic max on floats ----
__device__ __forceinline__ unsigned fenc(float f) {
  unsigned u = __float_as_uint(f);
  return (u & 0x80000000u) ? ~u : (u | 0x80000000u);
}
__device__ __forceinline__ float fdec(unsigned e) {
  unsigned u = (e & 0x80000000u) ? (e & 0x7FFFFFFFu) : ~e;
  return __uint_as_float(u);
}

// ---- K0a: M[r,h] = tokeys[r,h]^T @ toqueries[r,h]  (32x32 each) ----
__global__ void precompute_M_kernel(const float* __restrict__ tokeys,
                                    const float* __restrict__ toqueries,
                                    float* __restrict__ Mbuf) {
  int t = blockIdx.x * blockDim.x + threadIdx.x;  // 0..16383 = (rh*32+i)*32+j
  if (t >= RR * HH * SS * SS) return;
  int j  = t & 31;
  int i  = (t >> 5) & 31;
  int rh = t >> 10;
  const float* K = tokeys    + (size_t)rh * SS * SS;
  const float* Q = toqueries + (size_t)rh * SS * SS;
  float a = 0.f;
  #pragma unroll 8
  for (int k = 0; k < SS; ++k) a += K[k * SS + i] * Q[k * SS + j];
  Mbuf[t] = a;
}

// ---- K0b: Wk[(r*128+k)][i] = sum_{i2} unify[r,i,h*32+i2] * tovals[r,h,i2,j], k=h*32+j
__global__ void precompute_W_kernel(const float* __restrict__ unify,
                                    const float* __restrict__ tovals,
                                    float* __restrict__ Wk) {
  int t = blockIdx.x * blockDim.x + threadIdx.x;  // 0..65535 = kg*128 + i
  if (t >= RR * EMB * EMB) return;
  int i  = t & 127;
  int kg = t >> 7;          // 0..511
  int r  = kg >> 7;         // 0..3
  int k  = kg & 127;        // h*32 + j
  int h  = k >> 5, j = k & 31;
  const float* U = unify  + ((size_t)r * EMB + i) * EMB + h * SS;
  const float* V = tovals + (size_t)(r * HH + h) * SS * SS;
  float a = 0.f;
  #pragma unroll 8
  for (int i2 = 0; i2 < SS; ++i2) a += U[i2] * V[i2 * SS + j];
  Wk[t] = a;
}

// ---- K1: zero/seed workspace ----
__global__ void init_kernel(unsigned* __restrict__ segmax_u,
                            float* __restrict__ segsum,
                            float* __restrict__ acc) {
  int t = blockIdx.x * blockDim.x + threadIdx.x;
  const int nseg = NN * RR * HH;          // 800000
  const int nacc = NN * RR * EMB;         // 25600000
  if (t < nseg) {
    segmax_u[t] = fenc(-INFINITY);
    segsum[t] = 0.f;
  }
  if (t < nacc) acc[t] = 0.f;
}

// ---- K2: per-edge attention logits: dot[e,h] = xs^T M[r,h] xo  (wave per edge) ----
__global__ void edge_dot_kernel(const float* __restrict__ x,
                                const float* __restrict__ Mbuf,
                                const int* __restrict__ esub,
                                const int* __restrict__ epred,
                                const int* __restrict__ eobj,
                                float* __restrict__ dotbuf) {
  __shared__ float sxo[8][EMB];
  int wid = threadIdx.x >> 5, lane = threadIdx.x & 31;
  int e = blockIdx.x * 8 + wid;
  if (e >= EE) return;
  int s = esub[e], r = epred[e], o = eobj[e];
  #pragma unroll
  for (int h = 0; h < HH; ++h)
    sxo[wid][h * SS + lane] = x[(size_t)o * EMB + h * SS + lane];
  #pragma unroll
  for (int h = 0; h < HH; ++h) {
    float xs = x[(size_t)s * EMB + h * SS + lane];
    const float* Mrow = Mbuf + ((size_t)(r * HH + h) * SS + lane) * SS;
    float v = 0.f;
    #pragma unroll 8
    for (int j = 0; j < SS; ++j) v += Mrow[j] * sxo[wid][h * SS + j];
    float p = xs * v;
    #pragma unroll
    for (int off = 16; off > 0; off >>= 1) p += __shfl_down(p, off);
    if (lane == 0) dotbuf[(size_t)e * HH + h] = p;
  }
}

// ---- K3: segment max via atomic max on order-encoded uint ----
__global__ void segmax_kernel(const float* __restrict__ dotbuf,
                              const int* __restrict__ esub,
                              const int* __restrict__ epred,
                              unsigned* __restrict__ segmax_u) {
  int t = blockIdx.x * blockDim.x + threadIdx.x;
  if (t >= EE * HH) return;
  int e = t >> 2, h = t & 3;
  int s = esub[e], r = epred[e];
  atomicMax(&segmax_u[((size_t)s * RR + r) * HH + h], fenc(dotbuf[t]));
}

// ---- K4: exp(dot - max), segment sum ----
__global__ void expsum_kernel(float* __restrict__ dotbuf,
                              const int* __restrict__ esub,
                              const int* __restrict__ epred,
                              const unsigned* __restrict__ segmax_u,
                              float* __restrict__ segsum) {
  int t = blockIdx.x * blockDim.x + threadIdx.x;
  if (t >= EE * HH) return;
  int e = t >> 2, h = t & 3;
  int s = esub[e], r = epred[e];
  size_t si = ((size_t)s * RR + r) * HH + h;
  float ex = __expf(dotbuf[t] - fdec(segmax_u[si]));
  dotbuf[t] = ex;
  atomicAdd(&segsum[si], ex);
}

// ---- K5: scatter att * x[obj] into acc[n, r*128 + h*32 + j] (wave per edge) ----
__global__ void scatter_kernel(const float* __restrict__ x,
                               const float* __restrict__ dotbuf,
                               const float* __restrict__ segsum,
                               const int* __restrict__ esub,
                               const int* __restrict__ epred,
                               const int* __restrict__ eobj,
                               float* __restrict__ acc) {
  int wid = threadIdx.x >> 5, lane = threadIdx.x & 31;
  int e = blockIdx.x * 8 + wid;
  if (e >= EE) return;
  int s = esub[e], r = epred[e], o = eobj[e];
  size_t base = ((size_t)s * RR + r) * EMB;
  size_t segb = ((size_t)s * RR + r) * HH;
  #pragma unroll
  for (int h = 0; h < HH; ++h) {
    float att = dotbuf[(size_t)e * HH + h] / segsum[segb + h];
    float v = att * x[(size_t)o * EMB + h * SS + lane];
    atomicAdd(&acc[base + h * SS + lane], v);
  }
}

// ---- K6: out[N,128] = relu(acc[N,512] @ Wk[512,128]) with V_WMMA_F32_16X16X4_F32 ----
// One wave per 16x16 output tile; 8 waves/block cover 128 output columns.
__global__ void unify_wmma_kernel(const float* __restrict__ acc,
                                  const float* __restrict__ Wk,
                                  float* __restrict__ out) {
  const int rt    = blockIdx.x;          // row tile (rows rt*16 .. +15)
  const int wave  = threadIdx.x >> 5;    // col tile 0..7
  const int lane  = threadIdx.x & 31;
  const int lhalf = lane & 15;
  const int hi    = lane >> 4;           // 0: lanes 0-15, 1: lanes 16-31

  const int m0  = rt * 16;
  const int col = wave * 16 + lhalf;
  const float* arow = acc + (size_t)(m0 + lhalf) * (RR * EMB);

  v8f c = {};
  #pragma unroll 4
  for (int k0 = 0; k0 < RR * EMB; k0 += 4) {
    int ka = k0 + 2 * hi;
    // A 16x4 f32: VGPR0 = {K0 | K2}, VGPR1 = {K1 | K3} across the two lane halves
    v2f a; a.x = arow[ka];                 a.y = arow[ka + 1];
    // B 4x16 f32: rows striped across lanes (layout mirrors A with N on lanes)
    v2f b; b.x = Wk[(size_t)ka * EMB + col]; b.y = Wk[(size_t)(ka + 1) * EMB + col];
    c = __builtin_amdgcn_wmma_f32_16x16x4_f32(false, a, false, b,
                                              (short)0, c, false, false);
  }
  // C/D 16x16 f32: VGPR v -> M = v (lanes 0-15) / v+8 (lanes 16-31), N = lhalf
  #pragma unroll
  for (int v = 0; v < 8; ++v) {
    int m = m0 + v + hi * 8;
    float y = c[v];
    out[(size_t)m * EMB + wave * 16 + lhalf] = y > 0.f ? y : 0.f;
  }
}

extern "C" void kernel_launch(void* const* d_in, const int* in_sizes, int n_in,
                              void* d_out, int out_size, void* d_ws, size_t ws_size,
                              hipStream_t stream) {
  (void)in_sizes; (void)n_in; (void)out_size; (void)ws_size;
  const float* x         = (const float*)d_in[0];
  const float* tokeys    = (const float*)d_in[1];
  const float* toqueries = (const float*)d_in[2];
  const float* tovals    = (const float*)d_in[3];
  const float* unify     = (const float*)d_in[4];
  const int*   edge_sub  = (const int*)d_in[5];
  const int*   edge_pred = (const int*)d_in[6];
  const int*   edge_obj  = (const int*)d_in[7];
  float* out = (float*)d_out;

  // workspace carve-up (all 4-byte elements)
  float*    ws       = (float*)d_ws;
  float*    Mbuf     = ws;                                   // 16384
  float*    Wk       = Mbuf + RR * HH * SS * SS;             // 65536
  float*    dotbuf   = Wk + RR * EMB * EMB;                  // 1,600,000
  unsigned* segmax_u = (unsigned*)(dotbuf + (size_t)EE*HH);  // 800,000
  float*    segsum   = (float*)(segmax_u + (size_t)NN*RR*HH);// 800,000
  float*    acc      = segsum + (size_t)NN * RR * HH;        // 25,600,000

  precompute_M_kernel<<<(RR*HH*SS*SS + 255)/256, 256, 0, stream>>>(tokeys, toqueries, Mbuf);
  precompute_W_kernel<<<(RR*EMB*EMB + 255)/256, 256, 0, stream>>>(unify, tovals, Wk);
  init_kernel<<<(NN*RR*EMB + 255)/256, 256, 0, stream>>>(segmax_u, segsum, acc);
  edge_dot_kernel<<<EE/8, 256, 0, stream>>>(x, Mbuf, edge_sub, edge_pred, edge_obj, dotbuf);
  segmax_kernel<<<(EE*HH + 255)/256, 256, 0, stream>>>(dotbuf, edge_sub, edge_pred, segmax_u);
  expsum_kernel<<<(EE*HH + 255)/256, 256, 0, stream>>>(dotbuf, edge_sub, edge_pred, segmax_u, segsum);
  scatter_kernel<<<EE/8, 256, 0, stream>>>(x, dotbuf, segsum, edge_sub, edge_pred, edge_obj, acc);
  unify_wmma_kernel<<<NN/16, 256, 0, stream>>>(acc, Wk, out);
}